// GCNLayerPlaceholder_56779467653605
// MI455X (gfx1250) — compile-verified
//
#include <hip/hip_runtime.h>

typedef __attribute__((ext_vector_type(2))) float v2f;
typedef __attribute__((ext_vector_type(8))) float v8f;

#define GCN_F 64          // F_IN == F_OUT == 64

// ---------------------------------------------------------------------------
// Zero the accumulator (d_out) with float4 stores.
// ---------------------------------------------------------------------------
__global__ __launch_bounds__(256)
void gcn_zero_f4(float4* __restrict__ p, int n4) {
    int i = blockIdx.x * blockDim.x + threadIdx.x;
    if (i < n4) p[i] = make_float4(0.f, 0.f, 0.f, 0.f);
}

// ---------------------------------------------------------------------------
// transformed = H @ W via V_WMMA_F32_16X16X4_F32 (exact fp32).
// Block = 256 threads = 8 waves; W (16KB) staged in LDS once per block.
// Each wave computes a full 16x64 strip: 4 independent accumulator chains
// (ntile 0..3) so 4 WMMAs are in flight at once, and A-fragments are loaded
// exactly once per strip (no cross-wave redundancy on H).
// Frag layouts per ISA 7.12.2:
//   A 16x4:  lanes 0-15 -> K0(v0),K1(v1); lanes 16-31 -> K2(v0),K3(v1)
//   B 4x16:  mirrors A;  C/D: VGPR i -> rows i (lanes 0-15), i+8 (lanes 16-31)
// ---------------------------------------------------------------------------
__global__ __launch_bounds__(256)
void gcn_gemm_wmma_f32(const float* __restrict__ H,
                       const float* __restrict__ W,
                       float* __restrict__ T,
                       int n_rows) {
    __shared__ float Wl[GCN_F * GCN_F];          // 16 KB

    // Cooperative W load: 1024 float4 / 256 threads = 4 each (b128 vmem).
    {
        const float4* __restrict__ Wv = (const float4*)W;
        float4* Wlv = (float4*)Wl;
#pragma unroll
        for (int i = 0; i < 4; ++i)
            Wlv[threadIdx.x + 256 * i] = Wv[threadIdx.x + 256 * i];
    }
    __syncthreads();

    const int lane  = threadIdx.x & 31;
    const int wave  = threadIdx.x >> 5;          // 0..7
    const int mtile = blockIdx.x * 8 + wave;     // 16-row strip index

    if (mtile * 16 < n_rows) {                   // uniform per wave (EXEC all-1)
        const int half = lane >> 4;              // 0: lanes 0-15, 1: lanes 16-31
        const int nl   = lane & 15;
        const int m    = mtile * 16 + nl;        // row this lane supplies for A
        const float* __restrict__ hrow = H + (long long)m * GCN_F;

        v8f acc0 = {}, acc1 = {}, acc2 = {}, acc3 = {};
#pragma unroll
        for (int ks = 0; ks < 16; ++ks) {
            const int kbase = ks * 4 + half * 2;
            const v2f a = *(const v2f*)(hrow + kbase);       // b64 load

            v2f b0, b1, b2, b3;                              // ds_load x8
            b0.x = Wl[(kbase + 0) * GCN_F + nl +  0];
            b0.y = Wl[(kbase + 1) * GCN_F + nl +  0];
            b1.x = Wl[(kbase + 0) * GCN_F + nl + 16];
            b1.y = Wl[(kbase + 1) * GCN_F + nl + 16];
            b2.x = Wl[(kbase + 0) * GCN_F + nl + 32];
            b2.y = Wl[(kbase + 1) * GCN_F + nl + 32];
            b3.x = Wl[(kbase + 0) * GCN_F + nl + 48];
            b3.y = Wl[(kbase + 1) * GCN_F + nl + 48];

            acc0 = __builtin_amdgcn_wmma_f32_16x16x4_f32(false, a, false, b0,
                                                         (short)0, acc0, false, false);
            acc1 = __builtin_amdgcn_wmma_f32_16x16x4_f32(false, a, false, b1,
                                                         (short)0, acc1, false, false);
            acc2 = __builtin_amdgcn_wmma_f32_16x16x4_f32(false, a, false, b2,
                                                         (short)0, acc2, false, false);
            acc3 = __builtin_amdgcn_wmma_f32_16x16x4_f32(false, a, false, b3,
                                                         (short)0, acc3, false, false);
        }

        // Store C/D: VGPR i holds (row = mtile*16 + i + half*8, col).
#pragma unroll
        for (int i = 0; i < 8; ++i) {
            const int r = mtile * 16 + i + half * 8;
            float* __restrict__ trow = T + (long long)r * GCN_F + nl;
            trow[ 0] = acc0[i];
            trow[16] = acc1[i];
            trow[32] = acc2[i];
            trow[48] = acc3[i];
        }
    }
}

// ---------------------------------------------------------------------------
// COO SpMM scatter. Each wave owns 32 edges: every lane loads ONE edge's
// (row, col, val) once, then the metadata is broadcast with __shfl so the
// vmem stream is just b128 gathers + 4 no-return f32 atomics per lane.
// T (25.6MB) and out (25.6MB) are L2-resident (192MB L2).
// ---------------------------------------------------------------------------
__global__ __launch_bounds__(256)
void gcn_spmm_scatter(const int* __restrict__ erow,
                      const int* __restrict__ ecol,
                      const float* __restrict__ evals,
                      const float* __restrict__ T,
                      float* __restrict__ out,
                      int n_edges) {
    const int lane  = threadIdx.x & 31;
    const int wave  = threadIdx.x >> 5;
    const long long wgid = (long long)blockIdx.x * 8 + wave;
    const long long ebase = wgid * 32;
    if (ebase >= n_edges) return;                 // uniform per wave

    // One edge's metadata per lane.
    const long long eme = ebase + lane;
    const bool ok = (eme < n_edges);
    const int   rme = ok ? erow[eme]  : 0;
    const int   cme = ok ? ecol[eme]  : 0;
    const float vme = ok ? evals[eme] : 0.f;

    const int half = lane >> 4;                   // which of the 2 edges/iter
    const int f    = (lane & 15) << 2;            // feature chunk 0,4,...,60

#pragma unroll
    for (int i = 0; i < 16; ++i) {
        const int src = 2 * i + half;             // edge slot for this half-wave
        const int   r = __shfl(rme, src, 32);
        const int   c = __shfl(cme, src, 32);
        const float v = __shfl(vme, src, 32);
        if (ebase + src < n_edges) {
            const float4 t = *(const float4*)(T + (long long)c * GCN_F + f);
            float* o = out + (long long)r * GCN_F + f;
            atomicAdd(o + 0, v * t.x);            // global_atomic_add_f32
            atomicAdd(o + 1, v * t.y);
            atomicAdd(o + 2, v * t.z);
            atomicAdd(o + 3, v * t.w);
        }
    }
}

// ---------------------------------------------------------------------------
// out = relu(out + bias), in place, float4-vectorized.
// ---------------------------------------------------------------------------
__global__ __launch_bounds__(256)
void gcn_finalize(float* __restrict__ out, const float* __restrict__ bias, int n4) {
    int i = blockIdx.x * blockDim.x + threadIdx.x;
    if (i >= n4) return;
    float4 v = ((float4*)out)[i];
    const int fb = (i & ((GCN_F / 4) - 1)) << 2;          // (i*4) % 64
    const float4 b = *(const float4*)(bias + fb);
    v.x = fmaxf(v.x + b.x, 0.f);
    v.y = fmaxf(v.y + b.y, 0.f);
    v.z = fmaxf(v.z + b.z, 0.f);
    v.w = fmaxf(v.w + b.w, 0.f);
    ((float4*)out)[i] = v;
}

// ---------------------------------------------------------------------------
extern "C" void kernel_launch(void* const* d_in, const int* in_sizes, int n_in,
                              void* d_out, int out_size, void* d_ws, size_t ws_size,
                              hipStream_t stream) {
    const float* node_features = (const float*)d_in[0];   // [N, 64]
    const int*   edge_row      = (const int*)  d_in[1];   // [E]
    const int*   edge_col      = (const int*)  d_in[2];   // [E]
    const float* edge_vals     = (const float*)d_in[3];   // [E]
    const float* kern          = (const float*)d_in[4];   // [64, 64]
    const float* bias          = (const float*)d_in[5];   // [64]

    float* out = (float*)d_out;                            // [N, 64]
    float* T   = (float*)d_ws;                             // transformed [N, 64]

    const int n_nodes = in_sizes[0] / GCN_F;               // 100000
    const int n_edges = in_sizes[1];                       // 1600000

    // 1) zero the accumulator
    {
        const int n4 = (n_nodes * GCN_F) / 4;
        gcn_zero_f4<<<(n4 + 255) / 256, 256, 0, stream>>>((float4*)out, n4);
    }
    // 2) dense transform via WMMA f32 (block covers 128 rows; wave -> 16x64)
    {
        const int blocks = (n_nodes + 127) / 128;          // 782
        gcn_gemm_wmma_f32<<<blocks, 256, 0, stream>>>(node_features, kern, T, n_nodes);
    }
    // 3) edge scatter with f32 atomics (wave batches 32 edges)
    {
        const int blocks = (n_edges + 255) / 256;          // 6250 (8 waves/block)
        gcn_spmm_scatter<<<blocks, 256, 0, stream>>>(edge_row, edge_col, edge_vals,
                                                     T, out, n_edges);
    }
    // 4) bias + relu in place
    {
        const int n4 = (n_nodes * GCN_F) / 4;
        gcn_finalize<<<(n4 + 255) / 256, 256, 0, stream>>>(out, bias, n4);
    }
    (void)n_in; (void)out_size; (void)ws_size;
}